// FasterRCNN_78735340470369
// MI455X (gfx1250) — compile-verified
//
#include <hip/hip_runtime.h>
#include <hip/hip_bf16.h>

// ---------------------------------------------------------------------------
// Faster R-CNN RPN proposal pipeline for MI455X (gfx1250).
// Dominant op: 3x3 conv 512->512 on 64x64 (19.3 GFLOP) done as implicit GEMM
// with V_WMMA_F32_16X16X4_F32 (exact f32 to match the f32 reference).
// Round 3: K-major padded LDS layout so every WMMA fragment is one aligned
// ds_load_b64 (no v_mov reassembly), and 2x2 register blocking per wave
// (4 accumulators, 2 LDS dwords per WMMA). Block tile 64(M) x 128(N).
// ---------------------------------------------------------------------------

typedef __attribute__((ext_vector_type(2))) float v2f;
typedef __attribute__((ext_vector_type(8))) float v8f;

#define CONV_TM 64    // spatial tile (one full output row)
#define CONV_TN 128   // out-channel tile (2x2 sub-tiles per wave)
#define CONV_KC 32    // K chunk (input channels per stage)
#define CONV_KP 34    // padded K stride (even => 8B-aligned frags, no bank dup)

// ---------------------------------------------------------------------------
// Kernel 1: 3x3 conv, SAME padding, NCHW, via WMMA f32 16x16x4.
// Block: 256 threads = 8 waves. Wave (ms, ns) computes four 16x16 tiles:
// rows {ms*16, ms*16+32} x cols {ns*16, ns*16+64}. Grid: (64 rows, 4 n-tiles).
// ---------------------------------------------------------------------------
__global__ __launch_bounds__(256) void conv3x3_wmma(
    const float* __restrict__ feat,   // [512][64][64]
    const float* __restrict__ w,      // [512][512][3][3] (OIHW)
    const float* __restrict__ bias,   // [512]
    float* __restrict__ out)          // [512][64][64]
{
    __shared__ float As[CONV_TM][CONV_KP];   // [x][k]  8704 B
    __shared__ float Bs[CONV_TN][CONV_KP];   // [n][k] 17408 B

    const int tid  = threadIdx.x;
    const int wave = tid >> 5;
    const int lane = tid & 31;
    const int half = lane >> 4;     // 0: lanes 0-15, 1: lanes 16-31
    const int l15  = lane & 15;

    const int y     = blockIdx.x;               // output row
    const int nBase = blockIdx.y * CONV_TN;     // out-channel base

    const int ms = wave & 1;        // M sub-pair selector
    const int ns = wave >> 1;       // N sub-pair selector (0..3)
    const int m0 = ms * 16 + l15;   // A rows: m0 and m0+32
    const int n0 = ns * 16 + l15;   // B rows: n0 and n0+64

    v8f acc00 = {};  // (m0    , n0    )
    v8f acc01 = {};  // (m0    , n0+64 )
    v8f acc10 = {};  // (m0+32 , n0    )
    v8f acc11 = {};  // (m0+32 , n0+64 )

    for (int tap = 0; tap < 9; ++tap) {
        const int dy = tap / 3 - 1;
        const int dx = tap % 3 - 1;
        const int yy = y + dy;
        const bool rowOK = (yy >= 0) && (yy < 64);

        for (int c0 = 0; c0 < 512; c0 += CONV_KC) {
            // Stage A (K-major): As[x][k] = feature[c0+k][yy][x+dx].
            for (int t = tid; t < CONV_KC * CONV_TM; t += 256) {
                const int k  = t >> 6;          // 0..31
                const int x  = t & 63;          // coalesced over x
                const int xx = x + dx;
                float v = 0.0f;
                if (rowOK && (xx >= 0) && (xx < 64))
                    v = feat[(c0 + k) * 4096 + yy * 64 + xx];
                As[x][k] = v;
            }
            // Stage B (K-major): Bs[n][k] = w[nBase+n][c0+k][tap].
            for (int t = tid; t < CONV_KC * CONV_TN; t += 256) {
                const int k = t & 31;           // k fastest: small-stride gather
                const int n = t >> 5;
                Bs[n][k] = w[((nBase + n) * 512 + (c0 + k)) * 9 + tap];
            }
            // Prefetch next chunk's feature row (speculative, fault-safe).
            if (rowOK && (c0 + CONV_KC < 512)) {
                const float* nf =
                    feat + (c0 + CONV_KC + (tid >> 6)) * 4096 + yy * 64 + (tid & 63);
                __builtin_prefetch(nf, 0, 1);
            }
            __syncthreads();

            // 32 WMMAs over the 32-deep K chunk; each fragment = one b64 load.
            #pragma unroll
            for (int k = 0; k < CONV_KC; k += 4) {
                const int k2 = k + 2 * half;    // lanes 0-15: K={k,k+1}; 16-31: {k+2,k+3}
                const v2f a0 = *(const v2f*)(&As[m0     ][k2]);
                const v2f a1 = *(const v2f*)(&As[m0 + 32][k2]);
                const v2f b0 = *(const v2f*)(&Bs[n0     ][k2]);
                const v2f b1 = *(const v2f*)(&Bs[n0 + 64][k2]);
                acc00 = __builtin_amdgcn_wmma_f32_16x16x4_f32(
                    false, a0, false, b0, (short)0, acc00, false, false);
                acc01 = __builtin_amdgcn_wmma_f32_16x16x4_f32(
                    false, a0, false, b1, (short)0, acc01, false, false);
                acc10 = __builtin_amdgcn_wmma_f32_16x16x4_f32(
                    false, a1, false, b0, (short)0, acc10, false, false);
                acc11 = __builtin_amdgcn_wmma_f32_16x16x4_f32(
                    false, a1, false, b1, (short)0, acc11, false, false);
            }
            __syncthreads();
        }
    }

    // C/D layout: VGPR r -> M = r + 8*half (spatial x), N = l15 (out channel).
    {
        const int o0 = nBase + ns * 16 + l15;
        const int o1 = o0 + 64;
        const float bv0 = bias[o0];
        const float bv1 = bias[o1];
        #pragma unroll
        for (int r = 0; r < 8; ++r) {
            const int x0 = ms * 16 + r + 8 * half;
            const int x1 = x0 + 32;
            out[o0 * 4096 + y * 64 + x0] = acc00[r] + bv0;
            out[o1 * 4096 + y * 64 + x0] = acc01[r] + bv1;
            out[o0 * 4096 + y * 64 + x1] = acc10[r] + bv0;
            out[o1 * 4096 + y * 64 + x1] = acc11[r] + bv1;
        }
    }
}

// ---------------------------------------------------------------------------
// Kernel 2: the two 1x1 convs fused. o<18 -> cls_score, else bbox_pred.
// Grid (16, 54), block 256: thread computes one (channel, position) dot.
// ---------------------------------------------------------------------------
__global__ __launch_bounds__(256) void conv1x1(
    const float* __restrict__ rpn,
    const float* __restrict__ sw, const float* __restrict__ sb,
    const float* __restrict__ bw, const float* __restrict__ bb,
    float* __restrict__ out)          // [54][4096]
{
    const int pos = blockIdx.x * 256 + threadIdx.x;  // 0..4095
    const int o   = blockIdx.y;                      // 0..53
    const float* wv;
    float acc;
    if (o < 18) { wv = sw + o * 512;        acc = sb[o];      }
    else        { wv = bw + (o - 18) * 512; acc = bb[o - 18]; }
    for (int i = 0; i < 512; ++i)
        acc += rpn[i * 4096 + pos] * wv[i];
    out[o * 4096 + pos] = acc;
}

// ---------------------------------------------------------------------------
// Kernel 3: softmax + anchor decode + clip + min-size filter.
// Also initializes the (score, index) sort arrays padded to 65536.
// ---------------------------------------------------------------------------
__global__ __launch_bounds__(256) void decode_anchors(
    const float* __restrict__ convo,   // [54][4096]
    const float* __restrict__ im_info, // [3]
    float* __restrict__ proposals,     // [36864][4]
    float* __restrict__ key,           // [65536]
    int*   __restrict__ idx)           // [65536]
{
    const int t = blockIdx.x * 256 + threadIdx.x;
    if (t >= 65536) return;
    if (t >= 36864) { key[t] = -3.0e38f; idx[t] = t; return; }

    // Canonical base anchors (base 16, ratios .5/1/2, scales 8/16/32).
    const float base[9][4] = {
        { -84.f,  -40.f,  99.f,  55.f}, {-176.f,  -88.f, 191.f, 103.f},
        {-360.f, -184.f, 375.f, 199.f}, { -56.f,  -56.f,  71.f,  71.f},
        {-120.f, -120.f, 135.f, 135.f}, {-248.f, -248.f, 263.f, 263.f},
        { -36.f,  -80.f,  51.f,  95.f}, { -80.f, -168.f,  95.f, 183.f},
        {-168.f, -344.f, 183.f, 359.f}};

    const int pos = t / 9;
    const int a   = t - pos * 9;
    const int y   = pos >> 6;
    const int x   = pos & 63;

    // softmax over {bg, fg} -> fg prob.
    const float s_bg = convo[a * 4096 + pos];
    const float s_fg = convo[(9 + a) * 4096 + pos];
    const float fg   = 1.0f / (1.0f + __expf(s_bg - s_fg));

    const float d0 = convo[(18 + 4 * a + 0) * 4096 + pos];
    const float d1 = convo[(18 + 4 * a + 1) * 4096 + pos];
    const float d2 = convo[(18 + 4 * a + 2) * 4096 + pos];
    const float d3 = convo[(18 + 4 * a + 3) * 4096 + pos];

    const float ax1 = base[a][0] + x * 16.f;
    const float ay1 = base[a][1] + y * 16.f;
    const float ax2 = base[a][2] + x * 16.f;
    const float ay2 = base[a][3] + y * 16.f;
    const float aw  = ax2 - ax1 + 1.0f;
    const float ah  = ay2 - ay1 + 1.0f;
    const float acx = ax1 + 0.5f * aw;
    const float acy = ay1 + 0.5f * ah;

    const float pcx = d0 * aw + acx;
    const float pcy = d1 * ah + acy;
    const float pw  = __expf(d2) * aw;
    const float ph  = __expf(d3) * ah;

    const float im_h  = im_info[0];
    const float im_w  = im_info[1];
    const float scale = im_info[2];

    float px1 = fminf(fmaxf(pcx - 0.5f * pw, 0.0f), im_w - 1.0f);
    float py1 = fminf(fmaxf(pcy - 0.5f * ph, 0.0f), im_h - 1.0f);
    float px2 = fminf(fmaxf(pcx + 0.5f * pw, 0.0f), im_w - 1.0f);
    float py2 = fminf(fmaxf(pcy + 0.5f * ph, 0.0f), im_h - 1.0f);

    const float ws = px2 - px1 + 1.0f;
    const float hs = py2 - py1 + 1.0f;
    const bool valid = (ws >= 16.0f * scale) && (hs >= 16.0f * scale);

    proposals[t * 4 + 0] = px1;
    proposals[t * 4 + 1] = py1;
    proposals[t * 4 + 2] = px2;
    proposals[t * 4 + 3] = py2;
    key[t] = valid ? fg : -1.0f;
    idx[t] = t;
}

// ---------------------------------------------------------------------------
// Kernel 4: bitonic sort step over 65536 (score, index) pairs, descending by
// score with ascending-index tie-break (matches lax.top_k semantics).
// ---------------------------------------------------------------------------
__global__ __launch_bounds__(256) void bitonic_step(
    float* __restrict__ key, int* __restrict__ idx, int j, int k)
{
    const unsigned i = blockIdx.x * 256u + threadIdx.x;
    const unsigned p = i ^ (unsigned)j;
    if (p <= i) return;
    const float sa = key[i], sb = key[p];
    const int   ia = idx[i], ib = idx[p];
    const bool before   = (sa > sb) || ((sa == sb) && (ia < ib));
    const bool natural  = ((i & (unsigned)k) == 0);
    const bool doSwap   = natural ? !before : before;
    if (doSwap) {
        key[i] = sb; key[p] = sa;
        idx[i] = ib; idx[p] = ia;
    }
}

// ---------------------------------------------------------------------------
// Kernel 5: gather top-6000 proposals + their scores.
// ---------------------------------------------------------------------------
__global__ __launch_bounds__(256) void gather_props(
    const float* __restrict__ proposals,
    const float* __restrict__ key, const int* __restrict__ idx,
    float* __restrict__ props, float* __restrict__ topS)
{
    const int t = blockIdx.x * 256 + threadIdx.x;
    if (t >= 6000) return;
    const int g = idx[t];
    props[t * 4 + 0] = proposals[g * 4 + 0];
    props[t * 4 + 1] = proposals[g * 4 + 1];
    props[t * 4 + 2] = proposals[g * 4 + 2];
    props[t * 4 + 3] = proposals[g * 4 + 3];
    topS[t] = key[t];
}

// ---------------------------------------------------------------------------
// Kernel 6: NMS suppression bitmask: sup[i][jb] bit b set iff j = jb*32+b,
// j > i, IoU(i,j) > 0.7 (reference's +1 box convention).
// ---------------------------------------------------------------------------
#define NMS_N     6000
#define NMS_WORDS 188   // ceil(6000/32)

__global__ __launch_bounds__(256) void nms_sup(
    const float* __restrict__ props, unsigned* __restrict__ sup)
{
    const int t = blockIdx.x * 256 + threadIdx.x;
    if (t >= NMS_N * NMS_WORDS) return;
    const int i  = t / NMS_WORDS;
    const int jb = t - i * NMS_WORDS;

    const float x1 = props[i * 4 + 0], y1 = props[i * 4 + 1];
    const float x2 = props[i * 4 + 2], y2 = props[i * 4 + 3];
    const float ai = (x2 - x1 + 1.0f) * (y2 - y1 + 1.0f);

    unsigned m = 0;
    for (int b = 0; b < 32; ++b) {
        const int j = jb * 32 + b;
        if (j <= i || j >= NMS_N) continue;
        const float bx1 = props[j * 4 + 0], by1 = props[j * 4 + 1];
        const float bx2 = props[j * 4 + 2], by2 = props[j * 4 + 3];
        const float xx1 = fmaxf(x1, bx1), yy1 = fmaxf(y1, by1);
        const float xx2 = fminf(x2, bx2), yy2 = fminf(y2, by2);
        const float iw = fmaxf(xx2 - xx1 + 1.0f, 0.0f);
        const float ih = fmaxf(yy2 - yy1 + 1.0f, 0.0f);
        const float inter = iw * ih;
        const float aj = (bx2 - bx1 + 1.0f) * (by2 - by1 + 1.0f);
        const float iou = inter / (ai + aj - inter);
        if (iou > 0.7f) m |= (1u << b);
    }
    sup[i * NMS_WORDS + jb] = m;
}

// ---------------------------------------------------------------------------
// Kernel 7: sequential greedy pass (single block), exact reference semantics.
// ---------------------------------------------------------------------------
__global__ __launch_bounds__(256) void nms_greedy(
    const unsigned* __restrict__ sup, int* __restrict__ keep)
{
    __shared__ unsigned removed[NMS_WORDS];
    const int tid = threadIdx.x;
    for (int t = tid; t < NMS_WORDS; t += 256) removed[t] = 0;
    __syncthreads();
    for (int i = 0; i < NMS_N; ++i) {
        const bool alive = !((removed[i >> 5] >> (i & 31)) & 1u);
        __syncthreads();   // all read removed before anyone updates it
        if (alive) {
            for (int t = tid; t < NMS_WORDS; t += 256)
                removed[t] |= sup[i * NMS_WORDS + t];
        }
        if (tid == 0) keep[i] = alive ? 1 : 0;
        __syncthreads();
    }
}

// ---------------------------------------------------------------------------
// Kernel 8: final top-300: kept entries in order, then suppressed entries in
// index order with score -2.0 (replicates top_k(where(keep, s, -2), 300)).
// ---------------------------------------------------------------------------
__global__ void finalize_rois(
    const float* __restrict__ props, const float* __restrict__ topS,
    const int* __restrict__ keep, float* __restrict__ rois)
{
    if (threadIdx.x != 0 || blockIdx.x != 0) return;
    int cnt = 0;
    for (int t = 0; t < NMS_N && cnt < 300; ++t) {
        if (keep[t]) {
            rois[cnt * 6 + 0] = 0.0f;
            rois[cnt * 6 + 1] = props[t * 4 + 0];
            rois[cnt * 6 + 2] = props[t * 4 + 1];
            rois[cnt * 6 + 3] = props[t * 4 + 2];
            rois[cnt * 6 + 4] = props[t * 4 + 3];
            rois[cnt * 6 + 5] = topS[t];
            ++cnt;
        }
    }
    for (int t = 0; t < NMS_N && cnt < 300; ++t) {
        if (!keep[t]) {
            rois[cnt * 6 + 0] = 0.0f;
            rois[cnt * 6 + 1] = props[t * 4 + 0];
            rois[cnt * 6 + 2] = props[t * 4 + 1];
            rois[cnt * 6 + 3] = props[t * 4 + 2];
            rois[cnt * 6 + 4] = props[t * 4 + 3];
            rois[cnt * 6 + 5] = -2.0f;
            ++cnt;
        }
    }
}

// ---------------------------------------------------------------------------
// Host launcher. Inputs (setup_inputs order):
//   0 feature [1,512,64,64] f32   1 im_info [3] f32
//   2 conv_w [512,512,3,3]        3 conv_b [512]
//   4 score_w [18,512,1,1]        5 score_b [18]
//   6 bbox_w [36,512,1,1]         7 bbox_b [36]
// Output: rois [300,6] f32.
// ---------------------------------------------------------------------------
extern "C" void kernel_launch(void* const* d_in, const int* in_sizes, int n_in,
                              void* d_out, int out_size, void* d_ws, size_t ws_size,
                              hipStream_t stream) {
    const float* feature = (const float*)d_in[0];
    const float* im_info = (const float*)d_in[1];
    const float* conv_w  = (const float*)d_in[2];
    const float* conv_b  = (const float*)d_in[3];
    const float* score_w = (const float*)d_in[4];
    const float* score_b = (const float*)d_in[5];
    const float* bbox_w  = (const float*)d_in[6];
    const float* bbox_b  = (const float*)d_in[7];

    char* ws = (char*)d_ws;
    size_t off = 0;
    auto alloc = [&](size_t bytes) {
        void* p = ws + off;
        off += (bytes + 255) & ~(size_t)255;
        return p;
    };
    float*    rpn       = (float*)   alloc(512u * 4096u * 4u);      // 8 MB
    float*    convo     = (float*)   alloc(54u * 4096u * 4u);
    float*    proposals = (float*)   alloc(36864u * 4u * 4u);
    float*    key       = (float*)   alloc(65536u * 4u);
    int*      idx       = (int*)     alloc(65536u * 4u);
    float*    props     = (float*)   alloc(6000u * 4u * 4u);
    float*    topS      = (float*)   alloc(6000u * 4u);
    unsigned* sup       = (unsigned*)alloc((size_t)NMS_N * NMS_WORDS * 4u);
    int*      keep      = (int*)     alloc(6000u * 4u);
    (void)ws_size; (void)in_sizes; (void)n_in; (void)out_size;

    // 1) 3x3 conv via WMMA f32 (block tile 64x128, 4 accumulators per wave).
    conv3x3_wmma<<<dim3(64, 4), 256, 0, stream>>>(feature, conv_w, conv_b, rpn);

    // 2) fused 1x1 convs.
    conv1x1<<<dim3(16, 54), 256, 0, stream>>>(rpn, score_w, score_b, bbox_w, bbox_b, convo);

    // 3) softmax + decode + clip + filter; init sort arrays.
    decode_anchors<<<dim3(256), 256, 0, stream>>>(convo, im_info, proposals, key, idx);

    // 4) full bitonic sort of 65536 (score, index) pairs.
    for (int k = 2; k <= 65536; k <<= 1)
        for (int j = k >> 1; j > 0; j >>= 1)
            bitonic_step<<<dim3(65536 / 256), 256, 0, stream>>>(key, idx, j, k);

    // 5) gather top-6000.
    gather_props<<<dim3((6000 + 255) / 256), 256, 0, stream>>>(proposals, key, idx, props, topS);

    // 6) suppression bitmask + 7) greedy pass.
    nms_sup<<<dim3((NMS_N * NMS_WORDS + 255) / 256), 256, 0, stream>>>(props, sup);
    nms_greedy<<<dim3(1), 256, 0, stream>>>(sup, keep);

    // 8) final 300 rois.
    finalize_rois<<<dim3(1), 1, 0, stream>>>(props, topS, keep, (float*)d_out);
}